// HighOrder_24361054503519
// MI455X (gfx1250) — compile-verified
//
#include <hip/hip_runtime.h>

typedef __attribute__((ext_vector_type(2))) float v2f;
typedef __attribute__((ext_vector_type(8))) float v8f;

#define NOC   25      // conv output channels (= 5*5 taps)
#define NOCP  28      // padded stride -> 112B = 16B-aligned rows for b128 loads
#define KDIM  75      // fan-in: 3*5*5
#define KPAD  76      // pad K to multiple of 4
#define HALO  20      // 16 + 2*2
#define XSZ   (3*HALO*HALO)   // 1200

// compile-time patch offset: k -> c*400 + i*20 + j  (k >= 75 -> 0, weight is 0 there)
__device__ constexpr int patch_off(int k) {
    return (k < KDIM) ? (k / 25) * 400 + ((k % 25) / 5) * 20 + (k % 5) : 0;
}

__global__ __launch_bounds__(256)
void HighOrder_dynconv_fused(const float* __restrict__ x,   // [16,3,256,256]
                             const float* __restrict__ cw,  // [25,3,5,5] = [25,75]
                             const float* __restrict__ cb,  // [25]
                             float* __restrict__ out)       // [16,3,256,256]
{
    __shared__ float xs[XSZ];          // halo tile, [c][20][20]
    __shared__ float Wp[32 * KPAD];    // padded weights [32][76]
    __shared__ float biasL[NOC];
    __shared__ float wT[256 * NOCP];   // per-pixel dynamic weights (padded rows)

    const int tid = threadIdx.x;
    const int blk = blockIdx.x;
    const int b   = blk >> 8;          // 16 batches
    const int ty  = (blk >> 4) & 15;   // 16 tile rows
    const int tx  = blk & 15;          // 16 tile cols

    // ---- stage 0: cooperative LDS fill ----
    for (int idx = tid; idx < XSZ; idx += 256) {
        int c  = idx / 400;
        int r  = idx - c * 400;
        int ly = r / 20, lx = r - ly * 20;
        int gy = ty * 16 + ly - 2, gx = tx * 16 + lx - 2;
        float v = 0.0f;
        if ((unsigned)gy < 256u && (unsigned)gx < 256u)
            v = x[((b * 3 + c) * 256 + gy) * 256 + gx];
        xs[idx] = v;
    }
    for (int idx = tid; idx < 32 * KPAD; idx += 256) {
        int oc = idx / KPAD;
        int k  = idx - oc * KPAD;
        float v = 0.0f;
        if (oc < NOC && k < KDIM) v = cw[oc * KDIM + k];
        Wp[idx] = v;  // rows 25..31 and col 75 are zero padding
    }
    if (tid < NOC) biasL[tid] = cb[tid];
    __syncthreads();

    // ---- stage 1: per-pixel 25x75 GEMM via V_WMMA_F32_16X16X4_F32 ----
    const int lane  = tid & 31;
    const int wave  = tid >> 5;
    const int n     = lane & 15;          // pixel column within group / A row
    const int hi    = lane >> 4;          // upper lane half holds K+2, K+3
    const int khalf = hi << 1;

    for (int g = wave * 2; g < wave * 2 + 2; ++g) {   // two pixel rows per wave
        v8f acc0 = {};  // oc 0..15
        v8f acc1 = {};  // oc 16..31 (only 16..24 valid)
        const int base = g * 20 + n;   // patch origin in halo tile
        #pragma unroll
        for (int ch = 0; ch < 19; ++ch) {
            // constant-folded patch offsets, selected per lane half (no LDS table)
            const int o0 = hi ? patch_off(ch * 4 + 2) : patch_off(ch * 4 + 0);
            const int o1 = hi ? patch_off(ch * 4 + 3) : patch_off(ch * 4 + 1);
            v2f bfrag;
            bfrag.x = xs[o0 + base];
            bfrag.y = xs[o1 + base];
            const int k0 = ch * 4 + khalf;
            v2f a0 = *(const v2f*)(&Wp[n * KPAD + k0]);
            v2f a1 = *(const v2f*)(&Wp[(n + 16) * KPAD + k0]);
            acc0 = __builtin_amdgcn_wmma_f32_16x16x4_f32(
                       false, a0, false, bfrag, (short)0, acc0, false, false);
            acc1 = __builtin_amdgcn_wmma_f32_16x16x4_f32(
                       false, a1, false, bfrag, (short)0, acc1, false, false);
        }
        // spill dynamic weights (+bias) to LDS; D layout: VGPR r holds M=r / M=r+8
        const int pix    = g * 16 + n;
        const int ocbase = hi << 3;   // 0 or 8
        #pragma unroll
        for (int r = 0; r < 8; ++r) {
            int oc = ocbase + r;
            wT[pix * NOCP + oc] = acc0[r] + biasL[oc];
            int oc2 = 16 + ocbase + r;
            if (oc2 < NOC) wT[pix * NOCP + oc2] = acc1[r] + biasL[oc2];
        }
    }
    __syncthreads();

    // ---- stage 2: dynamic 5x5 aggregation + residual, one pixel per thread ----
    {
        const int py = tid >> 4, px = tid & 15;
        const int base = py * 20 + px;
        // 16B-aligned vector reads of this pixel's 25 weights (7 x ds_load_b128)
        float wloc[NOCP];
        const float4* wv = (const float4*)(&wT[tid * NOCP]);
        #pragma unroll
        for (int q = 0; q < 7; ++q) {
            float4 v = wv[q];
            wloc[q * 4 + 0] = v.x;
            wloc[q * 4 + 1] = v.y;
            wloc[q * 4 + 2] = v.z;
            wloc[q * 4 + 3] = v.w;
        }
        const int gy = ty * 16 + py, gx = tx * 16 + px;
        #pragma unroll
        for (int c = 0; c < 3; ++c) {
            float s = xs[c * 400 + (py + 2) * 20 + (px + 2)];  // residual x
            #pragma unroll
            for (int k = 0; k < NOC; ++k) {
                const int tap = (k / 5) * 20 + (k % 5);        // compile-time
                s += wloc[k] * xs[c * 400 + tap + base];
            }
            out[((b * 3 + c) * 256 + gy) * 256 + gx] = s;
        }
    }
}

extern "C" void kernel_launch(void* const* d_in, const int* in_sizes, int n_in,
                              void* d_out, int out_size, void* d_ws, size_t ws_size,
                              hipStream_t stream) {
    const float* x  = (const float*)d_in[0];   // 16*3*256*256
    const float* cw = (const float*)d_in[1];   // 25*75
    const float* cb = (const float*)d_in[2];   // 25
    float* out = (float*)d_out;
    // 16 batches * 16x16 tiles of 16x16 pixels
    dim3 grid(16 * 16 * 16);
    dim3 block(256);
    HighOrder_dynconv_fused<<<grid, block, 0, stream>>>(x, cw, cb, out);
}